// GConv_31817117729574
// MI455X (gfx1250) — compile-verified
//
#include <hip/hip_runtime.h>

// -------- CDNA5 WMMA fragment types (wave32) --------
typedef __attribute__((ext_vector_type(2))) float v2f;  // A/B frag: 16x4 or 4x16 f32 = 64 elem / 32 lanes
typedef __attribute__((ext_vector_type(8))) float v8f;  // C/D frag: 16x16 f32 = 256 elem / 32 lanes

// ---------------------------------------------------------------
// Kernel 1: zero the [N,160] accumulator living in workspace.
// ---------------------------------------------------------------
__global__ void GConv_zero_kernel(float* __restrict__ p, long n) {
    long i = (long)blockIdx.x * blockDim.x + threadIdx.x;
    long stride = (long)gridDim.x * blockDim.x;
    for (; i < n; i += stride) p[i] = 0.0f;
}

// ---------------------------------------------------------------
// Kernel 2: edge scatter. One wave32 per edge.
//   agg[dst][0:128]   += feat[src][0:128]   (4 atomics/lane, float4 load)
//   agg[dst][128:160] += edge_feat[e][0:32] (1 atomic/lane)
// agg (32MB) and feat (25.6MB) are L2-resident on MI455X (192MB L2),
// so these 160 atomics/edge resolve at L2 atomic units, not HBM.
// ---------------------------------------------------------------
__global__ void GConv_scatter_kernel(const float* __restrict__ feat,
                                     const float* __restrict__ efeat,
                                     const int* __restrict__ src,
                                     const int* __restrict__ dst,
                                     float* __restrict__ agg,
                                     int n_edges) {
    int e    = (int)((blockIdx.x * (long)blockDim.x + threadIdx.x) >> 5);
    int lane = threadIdx.x & 31;
    if (e >= n_edges) return;

    int s = src[e];
    int d = dst[e];

    // 128 floats of the source node row: lane handles floats [4*lane, 4*lane+3]
    const float4* frow = (const float4*)(feat + (long)s * 128);
    float4 v = frow[lane];                       // global_load_b128, coalesced

    float* arow = agg + (long)d * 160;
    atomicAdd(arow + 4 * lane + 0, v.x);         // global_atomic_add_f32 (no return)
    atomicAdd(arow + 4 * lane + 1, v.y);
    atomicAdd(arow + 4 * lane + 2, v.z);
    atomicAdd(arow + 4 * lane + 3, v.w);

    // 32 edge-feature floats: one per lane
    float ev = efeat[(long)e * 32 + lane];
    atomicAdd(arow + 128 + lane, ev);
}

// ---------------------------------------------------------------
// Kernel 3: fused GEMM + bias + residual using V_WMMA_F32_16X16X4_F32.
//   out[n][j] = feat[n][j] + b[j] + sum_k agg[n][k] * W[k][j]
// One wave computes one 16x16 output tile; 40 K-steps of 4.
//
// ISA 16x16x4 f32 layouts (CDNA5 ISA):
//   A (16x4):  lanes 0-15 -> M=lane,  VGPR0=K0, VGPR1=K1
//              lanes 16-31-> M=lane-16, VGPR0=K2, VGPR1=K3
//   B (4x16):  same striping with N across lanes
//   C (16x16): VGPR r: lanes 0-15 -> M=r, N=lane; lanes 16-31 -> M=8+r, N=lane-16
// ---------------------------------------------------------------
__global__ void GConv_wmma_gemm_kernel(const float* __restrict__ agg,   // [N,160]
                                       const float* __restrict__ W,     // [160,128]
                                       const float* __restrict__ bias,  // [128]
                                       const float* __restrict__ feat,  // [N,128]
                                       float* __restrict__ out,         // [N,128]
                                       int n_nodes) {
    const int waveId = threadIdx.x >> 5;
    const int lane   = threadIdx.x & 31;

    // 8 waves/block; waves in a block share m_tile (A-row reuse in L0/L2).
    const int tile   = blockIdx.x * 8 + waveId;
    const int m_tile = tile >> 3;
    const int n_tile = tile & 7;
    const int m0 = m_tile * 16;
    const int n0 = n_tile * 16;
    if (m0 >= n_nodes) return;   // wave-uniform: EXEC stays all-ones for WMMA

    const int half   = lane >> 4;    // 0: K pair (k0,k0+1), 1: (k0+2,k0+3)
    const int lanelo = lane & 15;

    const int rowA = m0 + lanelo;
    const int col  = n0 + lanelo;

    // A: agg[rowA][k0 + 2*half .. +1], 8-byte aligned -> float2 load
    const float* aptr = agg + (long)rowA * 160 + 2 * half;
    // B: W[k0 + 2*half][col] and next row
    const float* bptr = W + (long)(2 * half) * 128 + col;

    v8f c = {};
#pragma unroll
    for (int k0 = 0; k0 < 160; k0 += 4) {
        v2f a = *(const v2f*)(aptr + k0);        // global_load_b64
        v2f b;
        b.x = bptr[(long)k0 * 128];
        b.y = bptr[(long)(k0 + 1) * 128];
        // 8 args: (neg_a, A, neg_b, B, c_mod, C, reuse_a, reuse_b)
        c = __builtin_amdgcn_wmma_f32_16x16x4_f32(
                false, a, false, b, (short)0, c, false, false);
    }

    const float bb = bias[col];
#pragma unroll
    for (int r = 0; r < 8; ++r) {
        int row = m0 + r + 8 * half;
        long o  = (long)row * 128 + col;
        out[o] = feat[o] + bb + c[r];
    }
}

// ---------------------------------------------------------------
// Launch: zero -> scatter -> fused WMMA GEMM (stream-ordered).
// d_ws must hold N*160 floats (50000*160*4 = 32 MB).
// ---------------------------------------------------------------
extern "C" void kernel_launch(void* const* d_in, const int* in_sizes, int n_in,
                              void* d_out, int out_size, void* d_ws, size_t ws_size,
                              hipStream_t stream) {
    const float* feat  = (const float*)d_in[0];   // [N,128]
    const float* efeat = (const float*)d_in[1];   // [E,32]
    const int*   src   = (const int*)d_in[2];     // [E]
    const int*   dst   = (const int*)d_in[3];     // [E]
    const float* W     = (const float*)d_in[4];   // [160,128]
    const float* bias  = (const float*)d_in[5];   // [128]
    float*       out   = (float*)d_out;           // [N,128]

    const int n_nodes = in_sizes[0] / 128;
    const int n_edges = in_sizes[1] / 32;

    float* agg = (float*)d_ws;                    // [N,160] accumulator
    const long aggN = (long)n_nodes * 160;

    GConv_zero_kernel<<<4096, 256, 0, stream>>>(agg, aggN);

    const int scatterBlocks = (n_edges + 7) / 8;  // 8 waves (edges) per 256-thread block
    GConv_scatter_kernel<<<scatterBlocks, 256, 0, stream>>>(
        feat, efeat, src, dst, agg, n_edges);

    const int mtiles = (n_nodes + 15) / 16;       // 8 col-tiles handled by the 8 waves/block
    GConv_wmma_gemm_kernel<<<mtiles, 256, 0, stream>>>(
        agg, W, bias, feat, out, n_nodes);
}